// GATConv_65764539236418
// MI455X (gfx1250) — compile-verified
//
#include <hip/hip_runtime.h>

typedef __attribute__((ext_vector_type(2))) float v2f;
typedef __attribute__((ext_vector_type(8))) float v8f;

#define IN_DIM    256
#define OUT_DIM   128
#define LDS_PITCH 260   // 256 + 4 pad: row stride 1040B -> conflict-free A reads, 16B aligned
#define M_TILES   4     // 4 x 16 = 64 rows per block: each B load feeds 4 WMMAs

// ---------------------------------------------------------------------------
// Kernel 1: X_prime = X @ W   (fp32 WMMA, V_WMMA_F32_16X16X4_F32)
// Block = 256 threads = 8 waves. Block covers 64 rows of X; wave w owns the
// 16-wide N tile [w*16, w*16+16) across all 4 M-tiles (register tiling in M
// amortizes the B-operand loads: 1 dword of W per lane per 2 WMMAs).
// ---------------------------------------------------------------------------
__global__ __launch_bounds__(256)
void gat_proj_wmma(const float* __restrict__ X, const float* __restrict__ W,
                   float* __restrict__ Xp, int n_nodes) {
    __shared__ float sX[16 * M_TILES * LDS_PITCH];

    const int tid    = threadIdx.x;
    const int row_b  = blockIdx.x * (16 * M_TILES);

    // Cooperative stage of the 64x256 A block into LDS (16 float4 per thread).
    #pragma unroll
    for (int i = 0; i < 4 * M_TILES; ++i) {
        const int f = tid + i * 256;        // float4 slot in [0, 64*64)
        const int r = f >> 6;               // row   0..63
        const int c = (f & 63) << 2;        // col   0..252 (step 4)
        float4 v = make_float4(0.f, 0.f, 0.f, 0.f);
        if (row_b + r < n_nodes)
            v = *reinterpret_cast<const float4*>(X + (size_t)(row_b + r) * IN_DIM + c);
        *reinterpret_cast<float4*>(&sX[r * LDS_PITCH + c]) = v;
    }
    __syncthreads();

    const int lane  = tid & 31;
    const int wave  = tid >> 5;
    const int n0    = wave * 16;            // this wave's N tile
    const int mn    = lane & 15;            // M for A / N for B
    const int khalf = (lane >> 4) << 1;     // lanes 0-15 -> K+0,1 ; lanes 16-31 -> K+2,3

    const float* Wb = W + n0 + mn;          // column base for this lane's B elements

    v8f acc[M_TILES];
    #pragma unroll
    for (int t = 0; t < M_TILES; ++t) acc[t] = (v8f){};

    #pragma unroll 4
    for (int k = 0; k < IN_DIM; k += 4) {
        const int k0 = k + khalf;
        // B: W[k0][n0+mn], W[k0+1][n0+mn] — loaded once, reused by 4 WMMAs
        v2f b;
        b.x = Wb[(size_t)k0 * OUT_DIM];
        b.y = Wb[(size_t)(k0 + 1) * OUT_DIM];
        #pragma unroll
        for (int t = 0; t < M_TILES; ++t) {
            // A: X[row_b + t*16 + mn][k0..k0+1]  (8B-aligned LDS pair)
            v2f a = *reinterpret_cast<const v2f*>(&sX[(t * 16 + mn) * LDS_PITCH + k0]);
            acc[t] = __builtin_amdgcn_wmma_f32_16x16x4_f32(
                /*neg_a=*/false, a, /*neg_b=*/false, b,
                /*c_mod=*/(short)0, acc[t], /*reuse_a=*/false, /*reuse_b=*/false);
        }
    }

    // D layout: VGPR i -> M = i (lanes 0-15) / i+8 (lanes 16-31), N = lane%16.
    const int mbase = (lane >> 4) << 3;
    float* outp = Xp + n0 + mn;
    if (row_b + 16 * M_TILES <= n_nodes) {
        // fast path: full block, unguarded stores
        #pragma unroll
        for (int t = 0; t < M_TILES; ++t)
            #pragma unroll
            for (int i = 0; i < 8; ++i)
                outp[(size_t)(row_b + t * 16 + mbase + i) * OUT_DIM] = acc[t][i];
    } else {
        #pragma unroll
        for (int t = 0; t < M_TILES; ++t)
            #pragma unroll
            for (int i = 0; i < 8; ++i) {
                const int m = row_b + t * 16 + mbase + i;
                if (m < n_nodes) outp[(size_t)m * OUT_DIM] = acc[t][i];
            }
    }
}

// ---------------------------------------------------------------------------
// Kernel 2: fused SDDMM + attention scale + segment-sum SpMM.
// One wave per node (rows are contiguous CSR segments -> no atomics).
// Each lane holds 4 of the 128 output features; neighbor rows are gathered
// once (512B coalesced) and reused for both the dot product and the
// weighted accumulation. X_prime (51 MB) is L2-resident (192 MB L2).
// ---------------------------------------------------------------------------
__global__ __launch_bounds__(256)
void gat_edge_fused(const float* __restrict__ Xp, const float* __restrict__ attw,
                    const int* __restrict__ rowp, const int* __restrict__ colidx,
                    float* __restrict__ out, int n_nodes, int n_heads) {
    const int lane = threadIdx.x & 31;
    const int wave = threadIdx.x >> 5;
    const int node = blockIdx.x * 8 + wave;
    if (node >= n_nodes) return;

    float wsum = 0.f;                        // sum over heads of attention_w
    for (int h = 0; h < n_heads; ++h) wsum += attw[h];

    const float4 s = reinterpret_cast<const float4*>(Xp + (size_t)node * OUT_DIM)[lane];
    float4 acc = make_float4(0.f, 0.f, 0.f, 0.f);

    const int e0 = rowp[node];
    const int e1 = rowp[node + 1];
    #pragma unroll 4
    for (int e = e0; e < e1; ++e) {
        const int c = colidx[e];             // wave-uniform -> scalar load
        const float4 d = reinterpret_cast<const float4*>(Xp + (size_t)c * OUT_DIM)[lane];
        float p = s.x * d.x + s.y * d.y + s.z * d.z + s.w * d.w;
        // wave32 butterfly all-reduce of the 128-wide dot product
        #pragma unroll
        for (int m = 16; m >= 1; m >>= 1) p += __shfl_xor(p, m, 32);
        const float a = p * wsum;
        acc.x += a * d.x; acc.y += a * d.y; acc.z += a * d.z; acc.w += a * d.w;
    }

    reinterpret_cast<float4*>(out + (size_t)node * OUT_DIM)[lane] = acc;
}

// ---------------------------------------------------------------------------
extern "C" void kernel_launch(void* const* d_in, const int* in_sizes, int n_in,
                              void* d_out, int out_size, void* d_ws, size_t ws_size,
                              hipStream_t stream) {
    const float* X      = (const float*)d_in[0];
    const float* W      = (const float*)d_in[1];
    const float* attw   = (const float*)d_in[2];
    const int*   rowp   = (const int*)d_in[3];
    const int*   colidx = (const int*)d_in[4];
    // d_in[5..7]: blockPartition / edgeToColumn / edgeToRow — schedule metadata, no math effect.

    const int n_nodes = in_sizes[3] - 1;
    const int n_heads = in_sizes[2];

    float* Xp  = (float*)d_ws;    // [n_nodes, OUT_DIM] fp32 scratch
    float* out = (float*)d_out;

    const int gridA = (n_nodes + 16 * M_TILES - 1) / (16 * M_TILES);
    gat_proj_wmma<<<gridA, 256, 0, stream>>>(X, W, Xp, n_nodes);

    const int gridB = (n_nodes + 7) / 8;
    gat_edge_fused<<<gridB, 256, 0, stream>>>(Xp, attw, rowp, colidx, out, n_nodes, n_heads);
}